// FrequencyAwareGatingBlock_15753940042168
// MI455X (gfx1250) — compile-verified
//
#include <hip/hip_runtime.h>
#include <math.h>

// ---------------------------------------------------------------------------
// FrequencyAwareGatingBlock for MI455X (gfx1250), fp32 end-to-end.
//
// Folding: f5 = w_poly @ (freq_real * (w_expand @ f1))
//            = [ (w_poly * freq_real[col]) @ w_expand ] @ f1  =: W_comb @ f1
//          LN gamma folds into W_comb columns, beta folds into a bias vector.
// GEMMs use V_WMMA_F32_16X16X4_F32 (full-precision fp32 matrix path) with
// 2(M) x 4(N) register blocking per wave (0.75 ds_load_b64 per WMMA).
// W tiles are staged into LDS by the Tensor Data Mover (tensor_load_to_lds,
// one issue per workgroup per K-chunk, padded row stride 68 floats), waited
// with s_wait_tensorcnt; X tiles are staged manually (transpose + LN fuse).
// ---------------------------------------------------------------------------

#define CCH   256
#define C2CH  512
#define HWS   4096        // 64*64
#define NPOSN 65536       // B*H*W = 16*4096
#define EPSV  1e-5f

typedef __attribute__((ext_vector_type(2))) float v2f;
typedef __attribute__((ext_vector_type(8))) float v8f;
typedef __attribute__((ext_vector_type(4))) unsigned int u32x4;
typedef __attribute__((ext_vector_type(8))) int i32x8;
typedef __attribute__((ext_vector_type(4))) int i32x4;

// ---------------------------------------------------------------------------
// K0a: W_comb[o][c] = sum_k w_poly[o][k] * freq_real[k] * w_expand[k][c]
//      bias[o]      = sum_c W_comb[o][c] * ln_beta[c]      (pre-gamma)
//      Wcomb_out[o][c] = W_comb[o][c] * ln_gamma[c]
// ---------------------------------------------------------------------------
__global__ __launch_bounds__(256) void k_prep_w(
    const float* __restrict__ w_expand,   // [C2][C]
    const float* __restrict__ freq_real,  // [C2]
    const float* __restrict__ w_poly,     // [C2][C2]
    const float* __restrict__ ln_gamma,   // [C]
    const float* __restrict__ ln_beta,    // [C]
    float* __restrict__ Wcomb,            // [C2][C]
    float* __restrict__ bias)             // [C2]
{
  const int o = blockIdx.x;
  const int c = threadIdx.x;
  float acc = 0.f;
  for (int k = 0; k < C2CH; ++k)
    acc += w_poly[o * C2CH + k] * freq_real[k] * w_expand[k * CCH + c];

  __shared__ float red[256];
  red[c] = acc * ln_beta[c];
  __syncthreads();
  for (int s = 128; s > 0; s >>= 1) {
    if (c < s) red[c] += red[c + s];
    __syncthreads();
  }
  if (c == 0) bias[o] = red[0];
  Wcomb[o * CCH + c] = acc * ln_gamma[c];
}

// ---------------------------------------------------------------------------
// K0b: per-position LayerNorm statistics over the 256 input channels.
// ---------------------------------------------------------------------------
__global__ __launch_bounds__(256) void k_stats(
    const float* __restrict__ f, float* __restrict__ mu, float* __restrict__ rstd)
{
  const int n  = blockIdx.x * 256 + threadIdx.x;   // [0, 65536)
  const int b  = n >> 12;
  const int hw = n & 4095;
  const float* p = f + (size_t)b * (CCH * HWS) + hw;
  float s = 0.f, sq = 0.f;
  for (int c = 0; c < CCH; ++c) {
    float v = p[(size_t)c * HWS];
    s += v;
    sq += v * v;
  }
  const float m   = s * (1.0f / CCH);
  const float var = sq * (1.0f / CCH) - m * m;
  mu[n]   = m;
  rstd[n] = rsqrtf(var + EPSV);
}

// ---------------------------------------------------------------------------
// Depthwise 3x3, SAME (zero) padding, cross-correlation (lax.conv semantics).
// x,y layout: [C2][B][H][W] flat (idx = o*65536 + b*4096 + h*64 + w).
// ---------------------------------------------------------------------------
__global__ __launch_bounds__(256) void k_dw(
    const float* __restrict__ x, const float* __restrict__ w_dw, float* __restrict__ y)
{
  const size_t idx = (size_t)blockIdx.x * 256 + threadIdx.x;
  const int o   = (int)(idx >> 16);
  const int rem = (int)(idx & 65535u);
  const int hw  = rem & 4095;
  const int h   = hw >> 6;
  const int w   = hw & 63;
  const float* plane = x + (idx - (size_t)hw);   // start of [o][b] plane
  const float* k9 = w_dw + o * 9;
  float acc = 0.f;
#pragma unroll
  for (int dy = -1; dy <= 1; ++dy) {
    const int hh = h + dy;
    if ((unsigned)hh >= 64u) continue;
#pragma unroll
    for (int dx = -1; dx <= 1; ++dx) {
      const int ww = w + dx;
      if ((unsigned)ww >= 64u) continue;
      acc += k9[(dy + 1) * 3 + (dx + 1)] * plane[hh * 64 + ww];
    }
  }
  y[idx] = acc;
}

// ---------------------------------------------------------------------------
// TDM descriptor helper: 2D tile (tile0=64 f32 contiguous, tile1=256 rows,
// row stride = ktot elements) from global -> LDS with 4-DWORD padding after
// every 64 DWORDs (LDS row stride = 68 floats). All fields workgroup-uniform.
// D# layout per CDNA5 ISA 8.3/8.4; groups 2/3 (and trailing group) unused for
// a 2D non-gather, non-iterating tile -> zero-filled.
// ---------------------------------------------------------------------------
__device__ __forceinline__ void tdm_load_w_tile(const float* gsrc, unsigned lds_off,
                                                int ktot) {
  const unsigned long long ga = (unsigned long long)(uintptr_t)gsrc;
  u32x4 g0;
  g0[0] = 1u;                                        // count=1, user descriptor
  g0[1] = lds_off;                                   // lds_addr (bytes)
  g0[2] = (unsigned)(ga & 0xFFFFFFFFu);              // global_addr[31:0]
  g0[3] = (unsigned)((ga >> 32) & 0x01FFFFFFu)       // global_addr[56:32]
          | (2u << 30);                              // type=2 ("image")
  i32x8 g1;
  g1[0] = (2 << 16)                                  // data_size = 4B
          | (1 << 20)                                // pad_enable
          | (5 << 22)                                // pad_interval: every 64 DW
          | (3 << 25);                               // pad_amount: 4 DWORDs
  g1[1] = 64 << 16;                                  // tensor_dim0 = 64 (lo16)
  g1[2] = 256 << 16;                                 // dim0 hi / tensor_dim1=256 lo
  g1[3] = 64 << 16;                                  // dim1 hi / tile_dim0 = 64
  g1[4] = 256;                                       // tile_dim1 = 256, tile_dim2=0
  g1[5] = ktot;                                      // tensor_dim0_stride lo32
  g1[6] = 0;                                         // stride0 hi / stride1 lo
  g1[7] = 0;
  i32x4 z4 = {0, 0, 0, 0};
  i32x8 z8 = {0, 0, 0, 0, 0, 0, 0, 0};
  __builtin_amdgcn_tensor_load_to_lds(g0, g1, z4, z4, z8, 0);
}

// ---------------------------------------------------------------------------
// WMMA fp32 GEMM:  Y[M][N] = Wmat[M][KTOT] * X[KTOT][N] (+bias) (+epilogue)
//   MODE 0: X = input f (NCHW), normalized on the fly with mu/rstd;
//           accumulators init with bias; store to [M][NPOS] channel-major.
//   MODE 1: X = bufB [KTOT][NPOS]; epilogue f7 = gelu_exact(v)*v; store [M][NPOS].
//   MODE 2: X = bufA [KTOT][NPOS]; epilogue out = resid + v, NCHW layout.
//
// Block: 256 threads = 8 waves. Block tile: 256(M) x 64(N), K in 64-chunks.
// Wave register block: 2(M-tiles: rows w*16 and 128+w*16) x 4(N-tiles).
// LDS: W tile [256][68] via TDM (69.6KB) + X tile transposed [64][68] (17.4KB).
// A-fragment layout (f32 16x16x4): lane L holds row M=L%16,
//   K base = (L/16)*2 -> {W[m][k], W[m][k+1]} as one v2f. B symmetric.
// ---------------------------------------------------------------------------
template <int KTOT, int MODE>
__global__ __launch_bounds__(256) void k_gemm(
    const float* __restrict__ Wmat,
    const float* __restrict__ X,
    const float* __restrict__ mu,
    const float* __restrict__ rstd,
    const float* __restrict__ bias,
    const float* __restrict__ resid,
    float* __restrict__ Y)
{
  __shared__ float Wl[256 * 68];
  __shared__ float Xl[64 * 68];

  const int tid    = threadIdx.x;
  const int lane   = tid & 31;
  const int wave   = tid >> 5;
  const int m_base = blockIdx.y * 256;
  const int n_base = blockIdx.x * 64;
  const int halfhi = lane >> 4;          // 0 for lanes 0-15, 1 for lanes 16-31
  const unsigned wl_off = (unsigned)(uintptr_t)(void*)&Wl[0];  // LDS byte offset

  v8f acc[2][4];
#pragma unroll
  for (int mt = 0; mt < 2; ++mt) {
    if constexpr (MODE == 0) {
#pragma unroll
      for (int r = 0; r < 8; ++r) {
        const float bv = bias[m_base + mt * 128 + wave * 16 + (halfhi << 3) + r];
#pragma unroll
        for (int t = 0; t < 4; ++t) acc[mt][t][r] = bv;
      }
    } else {
#pragma unroll
      for (int t = 0; t < 4; ++t)
#pragma unroll
        for (int r = 0; r < 8; ++r) acc[mt][t][r] = 0.f;
    }
  }

  for (int k0 = 0; k0 < KTOT; k0 += 64) {
    // ---- W chunk via Tensor Data Mover: one issue per workgroup ----
    if (wave == 0)
      tdm_load_w_tile(Wmat + (size_t)m_base * KTOT + k0, wl_off, KTOT);

    // ---- stage X chunk: 64(K) x 64(N), transposed into Xl[n][k] ----
#pragma unroll
    for (int i = 0; i < 4; ++i) {
      const int id = tid + i * 256;       // 0..1023
      const int kr = id >> 4;             // 0..63
      const int n4 = (id & 15) << 2;      // 0..60
      const int gk = k0 + kr;
      const int gn = n_base + n4;
      float4 v;
      if constexpr (MODE == 0) {
        const int b  = gn >> 12;
        const int hw = gn & 4095;
        const float* gsrc = X + (size_t)b * (KTOT * HWS) + (size_t)gk * HWS + hw;
        v = *(const float4*)gsrc;
        if (k0 + 64 < KTOT) __builtin_prefetch(gsrc + (size_t)64 * HWS, 0, 1);
        v.x = (v.x - mu[gn + 0]) * rstd[gn + 0];
        v.y = (v.y - mu[gn + 1]) * rstd[gn + 1];
        v.z = (v.z - mu[gn + 2]) * rstd[gn + 2];
        v.w = (v.w - mu[gn + 3]) * rstd[gn + 3];
      } else {
        const float* gsrc = X + (size_t)gk * NPOSN + gn;
        v = *(const float4*)gsrc;
        if (k0 + 64 < KTOT) __builtin_prefetch(gsrc + (size_t)64 * NPOSN, 0, 1);
      }
      Xl[(n4 + 0) * 68 + kr] = v.x;
      Xl[(n4 + 1) * 68 + kr] = v.y;
      Xl[(n4 + 2) * 68 + kr] = v.z;
      Xl[(n4 + 3) * 68 + kr] = v.w;
    }
    if (wave == 0) __builtin_amdgcn_s_wait_tensorcnt(0);  // TDM done (wave 0)
    __syncthreads();                                      // orders all waves

    // ---- compute: 16 K-steps x (2 A + 4 B loads, 8 WMMAs) per wave ----
    const int mrow0 = wave * 16 + (lane & 15);
    const int mrow1 = 128 + mrow0;
    const int kb0   = halfhi << 1;
#pragma unroll
    for (int kk = 0; kk < 64; kk += 4) {
      const v2f a0 = *(const v2f*)&Wl[mrow0 * 68 + kk + kb0];
      const v2f a1 = *(const v2f*)&Wl[mrow1 * 68 + kk + kb0];
#pragma unroll
      for (int t = 0; t < 4; ++t) {
        const v2f bfr = *(const v2f*)&Xl[(t * 16 + (lane & 15)) * 68 + kk + kb0];
        acc[0][t] = __builtin_amdgcn_wmma_f32_16x16x4_f32(
            false, a0, false, bfr, (short)0, acc[0][t], false, false);
        acc[1][t] = __builtin_amdgcn_wmma_f32_16x16x4_f32(
            false, a1, false, bfr, (short)0, acc[1][t], false, false);
      }
    }
    __syncthreads();
  }

  // ---- epilogue / store ----
#pragma unroll
  for (int mt = 0; mt < 2; ++mt) {
#pragma unroll
    for (int t = 0; t < 4; ++t) {
#pragma unroll
      for (int r = 0; r < 8; ++r) {
        const int m = m_base + mt * 128 + wave * 16 + (halfhi << 3) + r;
        const int n = n_base + t * 16 + (lane & 15);
        const float v = acc[mt][t][r];
        if constexpr (MODE == 0) {
          Y[(size_t)m * NPOSN + n] = v;
        } else if constexpr (MODE == 1) {
          const float g = 0.5f * v * (1.0f + erff(v * 0.70710678118654752f));
          Y[(size_t)m * NPOSN + n] = g * v;       // gelu(v) * v
        } else {
          const int b  = n >> 12;
          const int hw = n & 4095;
          const size_t oi = (size_t)b * (CCH * HWS) + (size_t)m * HWS + hw;
          Y[oi] = resid[oi] + v;
        }
      }
    }
  }
}

// ---------------------------------------------------------------------------
extern "C" void kernel_launch(void* const* d_in, const int* in_sizes, int n_in,
                              void* d_out, int out_size, void* d_ws, size_t ws_size,
                              hipStream_t stream) {
  const float* f         = (const float*)d_in[0];
  const float* ln_gamma  = (const float*)d_in[1];
  const float* ln_beta   = (const float*)d_in[2];
  const float* w_expand  = (const float*)d_in[3];
  const float* freq_real = (const float*)d_in[4];
  // d_in[5] = freq_imag: contributes nothing (Re((r+ij)*x) = r*x for real x)
  const float* w_poly    = (const float*)d_in[6];
  const float* w_dw      = (const float*)d_in[7];
  const float* w_pw      = (const float*)d_in[8];
  const float* w_final   = (const float*)d_in[9];
  float* out = (float*)d_out;

  char* ws = (char*)d_ws;
  float* Wcomb = (float*)ws;  ws += (size_t)C2CH * CCH * 4;   // 512 KB
  float* bias  = (float*)ws;  ws += (size_t)C2CH * 4;         // 2 KB
  float* mu    = (float*)ws;  ws += (size_t)NPOSN * 4;        // 256 KB
  float* rstd  = (float*)ws;  ws += (size_t)NPOSN * 4;        // 256 KB
  float* bufA  = (float*)ws;  ws += (size_t)C2CH * NPOSN * 4; // 128 MB (f5 -> f7)
  float* bufB  = (float*)ws;                                  // 128 MB (f6a)

  // 1) fold poly/freq/expand + LN affine
  k_prep_w<<<C2CH, 256, 0, stream>>>(w_expand, freq_real, w_poly, ln_gamma, ln_beta,
                                     Wcomb, bias);
  // 2) LN statistics
  k_stats<<<NPOSN / 256, 256, 0, stream>>>(f, mu, rstd);
  // 3) f5 = Wcomb @ LN(f) + bias   (M=512, K=256, N=65536)
  {
    dim3 g(NPOSN / 64, C2CH / 256);
    k_gemm<CCH, 0><<<g, 256, 0, stream>>>(Wcomb, f, mu, rstd, bias, nullptr, bufA);
  }
  // 4) depthwise 3x3
  k_dw<<<(C2CH * NPOSN) / 256, 256, 0, stream>>>(bufA, w_dw, bufB);
  // 5) f7 = gelu(w_pw @ f6a) * (w_pw @ f6a)   (M=512, K=512)
  {
    dim3 g(NPOSN / 64, C2CH / 256);
    k_gemm<C2CH, 1><<<g, 256, 0, stream>>>(w_pw, bufB, nullptr, nullptr, nullptr, nullptr,
                                           bufA);
  }
  // 6) out = f + w_final @ f7   (M=256, K=512), NCHW output
  {
    dim3 g(NPOSN / 64, CCH / 256);
    k_gemm<C2CH, 2><<<g, 256, 0, stream>>>(w_final, bufA, nullptr, nullptr, nullptr, f,
                                           out);
  }
}